// AffinityPooling_74509092651251
// MI455X (gfx1250) — compile-verified
//
#include <hip/hip_runtime.h>

typedef __attribute__((ext_vector_type(2))) float f32x2;
typedef __attribute__((ext_vector_type(8))) float f32x8;

#define BDIM 8
#define NN   2048
#define DD   384
#define RR   64
#define HH   256
#define WW   256
#define EPSF 1e-8f

// ---- workspace layout (float indices) ----
#define OFF_KN     0                                  // 64*384      = 24576
#define OFF_S      24576                              // 8*2048*64   = 1048576
#define OFF_CHAT   (OFF_S + 1048576)                  // 1048576
#define OFF_COLSUM (OFF_CHAT + 1048576)               // 512
#define OFF_MASK   (OFF_COLSUM + 512)                 // 16384
#define OFF_ACC    (OFF_MASK + 16384)                 // 4
#define OFF_BITS   (OFF_ACC + 4)                      // 8*2048*64 u32 = 1048576
#define ZERO_START OFF_COLSUM
#define ZERO_COUNT (512 + 16384 + 4 + 1048576)

// ---- d_out layout: (out, segments, loss, C) concatenated ----
#define OUT_OFF_X    0
#define OUT_OFF_SEG  (BDIM * NN * DD)                 // 6291456
#define OUT_OFF_LOSS (OUT_OFF_SEG + BDIM * HH * WW)   // 6815744
#define OUT_OFF_C    (OUT_OFF_LOSS + 1)

// exact-precision f32 WMMA: D(16x16) = A(16x4) * B(4x16) + C
__device__ __forceinline__ f32x8 wmma_f32(f32x2 a, f32x2 b, f32x8 c) {
  return __builtin_amdgcn_wmma_f32_16x16x4_f32(false, a, false, b, (short)0, c,
                                               false, false);
}

__global__ void k_zero(float* ws) {
  size_t i = (size_t)blockIdx.x * blockDim.x + threadIdx.x;
  if (i < ZERO_COUNT) ws[ZERO_START + i] = 0.0f;
}

// normalize prototypes -> kn (R x D)
__global__ void k_protonorm(const float* __restrict__ protos, float* __restrict__ ws) {
  __shared__ float red[128];
  int r = blockIdx.x, t = threadIdx.x;
  const float* p = protos + r * DD;
  float s = 0.f;
  for (int d = t; d < DD; d += 128) { float v = p[d]; s += v * v; }
  red[t] = s; __syncthreads();
  for (int off = 64; off > 0; off >>= 1) { if (t < off) red[t] += red[t + off]; __syncthreads(); }
  float inv = 1.0f / fmaxf(sqrtf(red[0]), EPSF);
  float* kn = ws + OFF_KN + r * DD;
  for (int d = t; d < DD; d += 128) kn[d] = p[d] * inv;
}

// mask scatter + binary adjacency bitmask + float-cast segment output
__global__ void k_segments(const int* __restrict__ seg, float* __restrict__ ws,
                           float* __restrict__ outp) {
  int idx = blockIdx.x * blockDim.x + threadIdx.x;
  if (idx >= BDIM * HH * WW) return;
  int b = idx >> 16;
  int rem = idx & 65535;
  int y = rem >> 8, xx = rem & 255;
  int sraw = seg[idx];
  int s = sraw & (NN - 1);
  outp[OUT_OFF_SEG + idx] = (float)sraw;
  ws[OFF_MASK + b * NN + s] = 1.0f;
  unsigned* bits = (unsigned*)(ws + OFF_BITS);
  if (xx < WW - 1) {
    int s2 = seg[idx + 1] & (NN - 1);
    if (s2 != s && s != 0 && s2 != 0) {
      atomicOr(&bits[((size_t)b * NN + s) * (NN / 32) + (s2 >> 5)], 1u << (s2 & 31));
      atomicOr(&bits[((size_t)b * NN + s2) * (NN / 32) + (s >> 5)], 1u << (s & 31));
    }
  }
  if (y < HH - 1) {
    int s3 = seg[idx + WW] & (NN - 1);
    if (s3 != s && s != 0 && s3 != 0) {
      atomicOr(&bits[((size_t)b * NN + s) * (NN / 32) + (s3 >> 5)], 1u << (s3 & 31));
      atomicOr(&bits[((size_t)b * NN + s3) * (NN / 32) + (s >> 5)], 1u << (s & 31));
    }
  }
}

// S = rowsoftnorm((qn @ kn^T + 1)/2); also accumulate per-batch column sums of S.
// One block per (b, 16-node tile); 8 waves; WMMA over K=384 split 2 ways.
#define KNST 388
#define QNST 388
#define CSST 68
__global__ void __launch_bounds__(256) k_assign(const float* __restrict__ x,
                                                float* __restrict__ ws) {
  __shared__ float knl[RR * KNST];
  __shared__ float qnl[16 * QNST];
  __shared__ float csim[16 * CSST];
  __shared__ float red[256];
  __shared__ float normL[16];
  __shared__ float rs[16];
  __shared__ float colp[RR];

  int bid = blockIdx.x;
  int b = bid >> 7;
  int n0 = (bid & 127) * 16;
  int t = threadIdx.x;
  const float* kng = ws + OFF_KN;
  const float* xrow = x + ((size_t)b * NN + n0) * DD;

  for (int i = t; i < RR * DD; i += 256) knl[(i / DD) * KNST + (i % DD)] = kng[i];
  for (int i = t; i < 16 * DD; i += 256) qnl[(i / DD) * QNST + (i % DD)] = xrow[i];
  for (int i = t; i < 16 * RR; i += 256) csim[(i >> 6) * CSST + (i & 63)] = 0.0f;
  if (t < RR) colp[t] = 0.0f;
  __syncthreads();

  {  // row norms of the 16 x-rows
    int row = t >> 4, sub = t & 15;
    float acc = 0.f;
    for (int d = sub; d < DD; d += 16) { float v = qnl[row * QNST + d]; acc += v * v; }
    red[t] = acc;
  }
  __syncthreads();
  if ((t & 15) == 0) {
    int row = t >> 4;
    float s = 0.f;
    for (int k = 0; k < 16; ++k) s += red[row * 16 + k];
    normL[row] = 1.0f / fmaxf(sqrtf(s), EPSF);
  }
  __syncthreads();
  for (int i = t; i < 16 * DD; i += 256) qnl[(i / DD) * QNST + (i % DD)] *= normL[i / DD];
  __syncthreads();

  {  // WMMA: wave -> (r-tile = wv&3, K-half = wv>>2)
    int wv = t >> 5, lane = t & 31;
    int h = lane >> 4, lm = lane & 15;
    int nt = wv & 3, k0 = (wv >> 2) * 192;
    f32x8 acc = {};
    for (int kb = k0; kb < k0 + 192; kb += 4) {
      f32x2 a  = *(const f32x2*)&qnl[lm * QNST + kb + 2 * h];
      f32x2 bv = *(const f32x2*)&knl[(nt * 16 + lm) * KNST + kb + 2 * h];
      acc = wmma_f32(a, bv, acc);
    }
#pragma unroll
    for (int v = 0; v < 8; ++v)
      atomicAdd(&csim[(v + 8 * h) * CSST + nt * 16 + lm], acc[v]);
  }
  __syncthreads();

  for (int i = t; i < 16 * RR; i += 256) {
    int row = i >> 6, r = i & 63;
    csim[row * CSST + r] = (csim[row * CSST + r] + 1.0f) * 0.5f;
  }
  __syncthreads();
  if (t < 16) {
    float s = 0.f;
    for (int r = 0; r < RR; ++r) s += csim[t * CSST + r];
    rs[t] = s;
  }
  __syncthreads();
  float* Sg = ws + OFF_S + ((size_t)b * NN + n0) * RR;
  for (int i = t; i < 16 * RR; i += 256) {
    int row = i >> 6, r = i & 63;
    float sv = csim[row * CSST + r] / rs[row];
    Sg[row * RR + r] = sv;
    atomicAdd(&colp[r], sv);
  }
  __syncthreads();
  if (t < RR) atomicAdd(&ws[OFF_COLSUM + b * RR + t], colp[t]);
}

// per-node: DEC-KL loss terms, C = softmax(S), Chat = C/||C||
__global__ void k_node(float* __restrict__ ws, float* __restrict__ outp) {
  __shared__ float red[64];
  int bid = blockIdx.x;   // b*NN + n
  int b = bid >> 11;
  int t = threadIdx.x;
  float s = ws[OFF_S + (size_t)bid * RR + t];
  float cs = ws[OFF_COLSUM + b * RR + t];
  float pu = (s * s) / (cs + EPSF);

  red[t] = pu; __syncthreads();
  for (int off = 32; off > 0; off >>= 1) { if (t < off) red[t] += red[t + off]; __syncthreads(); }
  float rowP = red[0]; __syncthreads();

  float P = pu / (rowP + EPSF);
  float klt = P * (logf(P + EPSF) - logf(s + EPSF));
  red[t] = klt; __syncthreads();
  for (int off = 32; off > 0; off >>= 1) { if (t < off) red[t] += red[t + off]; __syncthreads(); }
  float kl = red[0]; __syncthreads();

  red[t] = s; __syncthreads();
  for (int off = 32; off > 0; off >>= 1) { if (t < off) red[t] = fmaxf(red[t], red[t + off]); __syncthreads(); }
  float mx = red[0]; __syncthreads();
  float e = __expf(s - mx);
  red[t] = e; __syncthreads();
  for (int off = 32; off > 0; off >>= 1) { if (t < off) red[t] += red[t + off]; __syncthreads(); }
  float se = red[0]; __syncthreads();
  float c = e / se;
  red[t] = c * c; __syncthreads();
  for (int off = 32; off > 0; off >>= 1) { if (t < off) red[t] += red[t + off]; __syncthreads(); }
  float cn = sqrtf(red[0]);

  outp[OUT_OFF_C + (size_t)bid * RR + t] = c;
  ws[OFF_CHAT + (size_t)bid * RR + t] = c / cn;
  if (t == 0) {
    float mk = ws[OFF_MASK + bid];
    atomicAdd(&ws[OFF_ACC + 0], kl * mk);
    atomicAdd(&ws[OFF_ACC + 1], mk);
  }
}

// Fused: w = mask(Chat @ Chat^T) row-panel in LDS (never hits HBM),
// then out = rownorm(w) @ x. Zero 16x16 tiles skipped via flags.
#define WST 2052   // 2052 % 64 == 4 -> conflict-free LDS row stride
#define AST 68
__global__ void __launch_bounds__(256) k_fused(const float* __restrict__ x,
                                               float* __restrict__ ws,
                                               float* __restrict__ outp) {
  __shared__ float wbuf[16 * WST];       // ~128 KB row panel
  __shared__ float atile[16 * AST];
  __shared__ float rowsum[16];
  __shared__ unsigned flags[128];

  int bid = blockIdx.x;
  int b = bid >> 7;
  int tileR = bid & 127;
  int n0 = tileR * 16;
  int t = threadIdx.x;
  int wv = t >> 5, lane = t & 31;
  int h = lane >> 4, lm = lane & 15;

  const float* chat = ws + OFF_CHAT + (size_t)b * NN * RR;
  const unsigned* bits = (const unsigned*)(ws + OFF_BITS) + (size_t)b * NN * (NN / 32);

  for (int i = t; i < 16 * RR; i += 256)
    atile[(i >> 6) * AST + (i & 63)] = chat[(size_t)(n0 + (i >> 6)) * RR + (i & 63)];
  if (t < 16) rowsum[t] = 0.0f;
  __syncthreads();

  float rsum[8];
#pragma unroll
  for (int v = 0; v < 8; ++v) rsum[v] = 0.0f;

  // phase 1: masked affinity panel, wave wv owns column tiles [wv*16, wv*16+16)
  for (int tt = 0; tt < 16; ++tt) {
    int m = wv * 16 + tt;
    unsigned word = bits[(size_t)(n0 + lm) * (NN / 32) + (m >> 1)];
    unsigned mrow = (word >> ((m & 1) * 16)) & 0xFFFFu;
    bool diagT = (m == tileR);
    unsigned long long bal = __ballot(mrow != 0u);
    bool any = ((bal & 0xFFFFull) != 0ull) || diagT;
    if (lane == 0) flags[m] = any ? 1u : 0u;
    if (!any) continue;                       // wave-uniform
    f32x8 acc = {};
    const float* crow = chat + (size_t)(m * 16 + lm) * RR;
#pragma unroll
    for (int kb = 0; kb < RR; kb += 4) {
      f32x2 a  = *(const f32x2*)&atile[lm * AST + kb + 2 * h];
      f32x2 bv = *(const f32x2*)&crow[kb + 2 * h];
      acc = wmma_f32(a, bv, acc);
    }
#pragma unroll
    for (int v = 0; v < 8; ++v) {
      int row = v + 8 * h;
      unsigned mr = __shfl(mrow, row);
      bool bit = (((mr >> lm) & 1u) != 0u) || (diagT && row == lm);
      float wval = bit ? acc[v] : 0.0f;
      wbuf[row * WST + m * 16 + lm] = wval;
      rsum[v] += wval;
    }
  }
#pragma unroll
  for (int v = 0; v < 8; ++v) atomicAdd(&rowsum[v + 8 * h], rsum[v]);
  __syncthreads();

  // phase 2: out tile = rownorm(w) @ x ; wave wv owns D-tiles wv, wv+8, wv+16
  f32x8 oacc[3] = {};
  const float* xb = x + (size_t)b * NN * DD;
  for (int m = 0; m < 128; ++m) {
    if (flags[m] == 0u) continue;             // block-uniform
#pragma unroll
    for (int kk = 0; kk < 4; ++kk) {
      int kg = m * 16 + kk * 4 + 2 * h;
      f32x2 a = *(const f32x2*)&wbuf[lm * WST + kg];
#pragma unroll
      for (int q = 0; q < 3; ++q) {
        int d0 = (wv + 8 * q) * 16;
        const float* xk = xb + (size_t)kg * DD + d0 + lm;
        f32x2 bv;
        bv.x = xk[0];
        bv.y = xk[DD];
        oacc[q] = wmma_f32(a, bv, oacc[q]);
      }
    }
  }
#pragma unroll
  for (int q = 0; q < 3; ++q) {
    int d0 = (wv + 8 * q) * 16;
#pragma unroll
    for (int v = 0; v < 8; ++v) {
      int row = v + 8 * h;
      float inv = 1.0f / (rowsum[row] + EPSF);
      outp[OUT_OFF_X + ((size_t)b * NN + n0 + row) * DD + d0 + lm] = oacc[q][v] * inv;
    }
  }
}

__global__ void k_loss(const float* __restrict__ ws, float* __restrict__ outp) {
  outp[OUT_OFF_LOSS] = ws[OFF_ACC + 0] / (ws[OFF_ACC + 1] + EPSF);
}

extern "C" void kernel_launch(void* const* d_in, const int* in_sizes, int n_in,
                              void* d_out, int out_size, void* d_ws, size_t ws_size,
                              hipStream_t stream) {
  const float* x      = (const float*)d_in[0];
  const int*   seg    = (const int*)d_in[1];
  const float* protos = (const float*)d_in[2];
  float* outp = (float*)d_out;
  float* ws   = (float*)d_ws;
  (void)in_sizes; (void)n_in; (void)out_size; (void)ws_size;

  k_zero<<<(ZERO_COUNT + 255) / 256, 256, 0, stream>>>(ws);
  k_protonorm<<<RR, 128, 0, stream>>>(protos, ws);
  k_segments<<<(BDIM * HH * WW + 255) / 256, 256, 0, stream>>>(seg, ws, outp);
  k_assign<<<BDIM * (NN / 16), 256, 0, stream>>>(x, ws);
  k_node<<<BDIM * NN, RR, 0, stream>>>(ws, outp);
  k_fused<<<BDIM * (NN / 16), 256, 0, stream>>>(x, ws, outp);
  k_loss<<<1, 1, 0, stream>>>(ws, outp);
}